// DeformConv2d_41205916238501
// MI455X (gfx1250) — compile-verified
//
#include <hip/hip_runtime.h>
#include <math.h>

typedef __attribute__((ext_vector_type(2))) float v2f;
typedef __attribute__((ext_vector_type(8))) float v8f;
typedef __attribute__((ext_vector_type(4))) unsigned int u32x4;
typedef __attribute__((ext_vector_type(4))) int i32x4;
typedef __attribute__((ext_vector_type(8))) int i32x8;

#define BATCH 2
#define CIN   256
#define COUT  256
#define Hh    96
#define Ww    96
#define HW    (Hh*Ww)          // 9216
#define NPOS  (BATCH*HW)       // 18432
#define NK    9
#define KTOT  (CIN*NK)         // 2304
#define HP    98               // padded extent

struct Samp { int4 q; float4 g; };   // 32 B: 4 gather indices + 4 bilinear weights

// ---------------------------------------------------------------------------
// Kernel 1: offset conv (18 ch) + ad conv (3 ch) + sampling prep
// ---------------------------------------------------------------------------
#define ICB1 32

__global__ __launch_bounds__(256)
void deform_prep(const float* __restrict__ x,
                 const float* __restrict__ w_p, const float* __restrict__ b_p,
                 const float* __restrict__ w_ad, const float* __restrict__ b_ad,
                 Samp* __restrict__ samp)
{
    __shared__ float wlds[21 * ICB1 * 9];    // 6048 floats = 24.2 KB

    const int t  = threadIdx.x;
    const int P  = blockIdx.x * 256 + t;
    const int b  = P / HW;
    const int pp = P - b * HW;
    const int h  = pp / Ww;
    const int w  = pp - h * Ww;

    float acc[21];
#pragma unroll
    for (int o = 0; o < 18; ++o) acc[o] = b_p[o];
#pragma unroll
    for (int o = 0; o < 3; ++o) acc[18 + o] = b_ad[o];

    const float* xb = x + (size_t)b * CIN * HW;

    for (int ic0 = 0; ic0 < CIN; ic0 += ICB1) {
        __syncthreads();
        for (int e = t; e < 21 * ICB1 * 9; e += 256) {
            int o   = e / (ICB1 * 9);
            int rem = e - o * (ICB1 * 9);
            wlds[e] = (o < 18) ? w_p[o * KTOT + ic0 * 9 + rem]
                               : w_ad[(o - 18) * KTOT + ic0 * 9 + rem];
        }
        __syncthreads();

        for (int icl = 0; icl < ICB1; ++icl) {
            const float* xc = xb + (size_t)(ic0 + icl) * HW;
            float v[9];
#pragma unroll
            for (int ky = 0; ky < 3; ++ky)
#pragma unroll
                for (int kx = 0; kx < 3; ++kx) {
                    int yy = h + ky - 1, xx = w + kx - 1;
                    v[ky * 3 + kx] = (yy >= 0 && yy < Hh && xx >= 0 && xx < Ww)
                                         ? xc[yy * Ww + xx] : 0.f;
                }
#pragma unroll
            for (int o = 0; o < 21; ++o) {
                const float* wr = &wlds[o * (ICB1 * 9) + icl * 9];
#pragma unroll
                for (int tt = 0; tt < 9; ++tt)
                    acc[o] = fmaf(wr[tt], v[tt], acc[o]);
            }
        }
    }

    // ---- sampling-position math (mirrors reference ordering exactly) ----
#pragma unroll
    for (int k = 0; k < 9; ++k) {
        float pnx = (float)(k / 3) - 1.f;
        float pny = (float)(k % 3) - 1.f;
        float ad  = 2.f / (1.f + __expf(acc[18 + (k % 3)]));   // DIL*(1-sigmoid)

        float px = (float)(h + 1) + pnx + acc[k]     + ad * pnx;
        float py = (float)(w + 1) + pny + acc[9 + k] + ad * pny;

        float fx = floorf(px), fy = floorf(py);
        int qltx = min(max((int)fx, 0), HP - 1);
        int qlty = min(max((int)fy, 0), HP - 1);
        int qrbx = min(max((int)fx + 1, 0), HP - 1);
        int qrby = min(max((int)fy + 1, 0), HP - 1);

        bool mx = (px < 1.f) || (px > (float)(HP - 2));
        bool my = (py < 1.f) || (py > (float)(HP - 2));
        if (mx) px = fx;
        if (my) py = fy;
        px = fminf(fmaxf(px, 0.f), (float)(HP - 1));
        py = fminf(fmaxf(py, 0.f), (float)(HP - 1));

        float glt = (1.f + ((float)qltx - px)) * (1.f + ((float)qlty - py));
        float grb = (1.f - ((float)qrbx - px)) * (1.f - ((float)qrby - py));
        float glb = (1.f + ((float)qltx - px)) * (1.f - ((float)qrby - py));
        float grt = (1.f - ((float)qrbx - px)) * (1.f + ((float)qlty - py));

        int4 q; float4 g;
        {
            bool ok = (qltx >= 1) && (qltx <= Hh) && (qlty >= 1) && (qlty <= Ww);
            q.x = ok ? ((qltx - 1) * Ww + (qlty - 1)) : 0;  g.x = ok ? glt : 0.f;
        }
        {
            bool ok = (qrbx >= 1) && (qrbx <= Hh) && (qrby >= 1) && (qrby <= Ww);
            q.y = ok ? ((qrbx - 1) * Ww + (qrby - 1)) : 0;  g.y = ok ? grb : 0.f;
        }
        {
            bool ok = (qltx >= 1) && (qltx <= Hh) && (qrby >= 1) && (qrby <= Ww);
            q.z = ok ? ((qltx - 1) * Ww + (qrby - 1)) : 0;  g.z = ok ? glb : 0.f;
        }
        {
            bool ok = (qrbx >= 1) && (qrbx <= Hh) && (qlty >= 1) && (qlty <= Ww);
            q.w = ok ? ((qrbx - 1) * Ww + (qlty - 1)) : 0;  g.w = ok ? grt : 0.f;
        }
        Samp s; s.q = q; s.g = g;
        samp[(size_t)P * NK + k] = s;
    }
}

// ---------------------------------------------------------------------------
// Kernel 2: fused bilinear-gather + implicit GEMM via V_WMMA_F32_16X16X4_F32
//   out[b,oc,p] = sum_{ic,k} w_conv[oc, ic*9+k] * sampled[b, ic, p, k]
//   M = 256 (oc), N = 18432 (b*h*w), K = 2304
//   A tile (128 x 64 f32, row stride 2304) staged via Tensor Data Mover with
//   LDS padding 64dw+4dw -> stride 68 floats (conflict-free ds_load_b64).
// ---------------------------------------------------------------------------
#define OCT  128      // oc tile per block
#define PT   64       // pos tile per block
#define KC   64       // K columns staged per iteration
#define ASTR 68       // 64 data + 4 pad DWORDs per row (TDM pad)
#define BSTR 68

__device__ __forceinline__ void tdm_load_A(unsigned lds_off, const float* gptr)
{
    unsigned long long ga = (unsigned long long)(uintptr_t)gptr;
    // D# group 0: count=1, lds_addr, global_addr[56:0], type=2
    u32x4 g0 = { 1u,
                 lds_off,
                 (unsigned)ga,
                 (unsigned)(ga >> 32) | (2u << 30) };
    // D# group 1: data_size=4B(2), pad_enable, pad_interval=64dw(5),
    //             pad_amount=4dw(3); tensor_dim0=2304, tensor_dim1=128,
    //             tile_dim0=64, tile_dim1=128, tensor_dim0_stride=2304
    i32x8 g1 = { (int)((2u << 16) | (1u << 20) | (5u << 22) | (3u << 25)),
                 (int)((KTOT & 0xFFFFu) << 16),
                 (int)(((unsigned)OCT & 0xFFFFu) << 16),
                 (int)(((unsigned)KC) << 16),
                 (int)OCT,
                 (int)KTOT,
                 0, 0 };
    i32x4 z4 = { 0, 0, 0, 0 };
#if __has_include(<hip/amd_detail/amd_gfx1250_TDM.h>)
    i32x8 z8 = { 0, 0, 0, 0, 0, 0, 0, 0 };
    __builtin_amdgcn_tensor_load_to_lds(g0, g1, z4, z4, z8, 0);
#else
    __builtin_amdgcn_tensor_load_to_lds(g0, g1, z4, z4, 0);
#endif
}

__global__ __launch_bounds__(256)
void deform_gemm(const float* __restrict__ x,
                 const float* __restrict__ w_conv,
                 const Samp* __restrict__ samp,
                 float* __restrict__ out)
{
    __shared__ float Alds[OCT * ASTR];   // 34.8 KB  (row = oc, col = K, TDM-filled)
    __shared__ float Blds[PT * BSTR];    // 17.4 KB  (row = pos, col = K)

    const int t    = threadIdx.x;
    const int lane = t & 31;
    const int wid  = t >> 5;
    const int wo   = wid & 3;            // oc  sub-tile group (32 oc each)
    const int wp   = wid >> 2;           // pos sub-tile group (32 pos each)
    const int pos0 = blockIdx.x * PT;
    const int oc0  = blockIdx.y * OCT;
    const int bimg = pos0 / HW;          // HW % PT == 0 -> tile within one image
    const float* xb = x + (size_t)bimg * CIN * HW;

    const int m  = lane & 15;            // row/col-in-tile
    const int kh = lane >> 4;            // K-half select (ISA lane layout)
    const unsigned ldsA = (unsigned)(uintptr_t)(void*)Alds;

    v8f accv[2][2] = {};

    for (int it = 0; it < KTOT / KC; ++it) {
        __syncthreads();                 // previous chunk fully consumed

        // ---- A tile via Tensor Data Mover (wave 0 issues, others gather) ----
        if (wid == 0)
            tdm_load_A(ldsA, w_conv + (size_t)oc0 * KTOT + it * KC);

        // ---- build B: 64 pos x 64 K via 4-point bilinear gather ----
        const int kbase = it * KC;
        for (int j = 0; j < 16; ++j) {
            int e  = t + 256 * j;
            int kk = e >> 6;             // 0..63  (K column in chunk)
            int pl = e & 63;             // consecutive lanes -> consecutive pos
            int gk = kbase + kk;         // global K index = ic*9 + k
            int ici = gk / 9;
            int k   = gk - ici * 9;
            Samp s = samp[(size_t)(pos0 + pl) * NK + k];
            const float* xc = xb + (size_t)ici * HW;
            float val = s.g.x * xc[s.q.x] + s.g.y * xc[s.q.y]
                      + s.g.z * xc[s.q.z] + s.g.w * xc[s.q.w];
            Blds[pl * BSTR + kk] = val;
        }

        if (wid == 0)
            __builtin_amdgcn_s_wait_tensorcnt(0);   // A tile landed in LDS
        __syncthreads();                            // publish A + B to all waves

        // ---- 16 K-steps of 16x16x4 fp32 WMMA, 2x2 tiles per wave ----
#pragma unroll
        for (int ks = 0; ks < KC / 4; ++ks) {
            const int kc = ks * 4 + kh * 2;
            v2f a[2], bf[2];
#pragma unroll
            for (int i = 0; i < 2; ++i)
                a[i] = *(const v2f*)&Alds[(wo * 32 + i * 16 + m) * ASTR + kc];
#pragma unroll
            for (int j = 0; j < 2; ++j)
                bf[j] = *(const v2f*)&Blds[(wp * 32 + j * 16 + m) * BSTR + kc];
#pragma unroll
            for (int i = 0; i < 2; ++i)
#pragma unroll
                for (int j = 0; j < 2; ++j)
                    accv[i][j] = __builtin_amdgcn_wmma_f32_16x16x4_f32(
                        false, a[i], false, bf[j],
                        (short)0, accv[i][j], false, false);
        }
    }

    // ---- epilogue: C/D layout -> out[b, oc, p] ----
#pragma unroll
    for (int i = 0; i < 2; ++i) {
        int oc_base = oc0 + wo * 32 + i * 16 + kh * 8;
#pragma unroll
        for (int j = 0; j < 2; ++j) {
            int P  = pos0 + wp * 32 + j * 16 + m;
            int pp = P - bimg * HW;
            float* op = out + (size_t)bimg * COUT * HW + pp;
#pragma unroll
            for (int r = 0; r < 8; ++r)
                op[(size_t)(oc_base + r) * HW] = accv[i][j][r];
        }
    }
}

// ---------------------------------------------------------------------------
extern "C" void kernel_launch(void* const* d_in, const int* in_sizes, int n_in,
                              void* d_out, int out_size, void* d_ws, size_t ws_size,
                              hipStream_t stream) {
    const float* x      = (const float*)d_in[0];
    const float* w_p    = (const float*)d_in[1];
    const float* b_p    = (const float*)d_in[2];
    const float* w_ad   = (const float*)d_in[3];
    const float* b_ad   = (const float*)d_in[4];
    const float* w_conv = (const float*)d_in[5];
    float* out = (float*)d_out;
    Samp* samp = (Samp*)d_ws;            // 18432*9*32 B = 5.3 MB

    deform_prep<<<NPOS / 256, 256, 0, stream>>>(x, w_p, b_p, w_ad, b_ad, samp);

    dim3 grid(NPOS / PT, COUT / OCT);    // 288 x 2
    deform_gemm<<<grid, 256, 0, stream>>>(x, w_conv, samp, out);
}